// ConvolutionalModel_43138651521330
// MI455X (gfx1250) — compile-verified
//
#include <hip/hip_runtime.h>

typedef __attribute__((ext_vector_type(2))) float v2f;
typedef __attribute__((ext_vector_type(8))) float v8f;

#define NH 127
#define NW 127
#define NBATCH 8
#define NWIN (NBATCH * NH * NW)   // 129032
#define HW (1024 * 1024)
#define LSTRIDE 260               // 256 + 4 pad: conflict-free A loads / C stores
#define MTILE 32                  // windows per block (2 WMMA M-tiles)

// out[32x256] = in[32x256] @ W[256x256] + bias, optional ReLU.
// Each wave: 2 M-tiles (rows 0-15, 16-31) x 2 N-tiles (wave, wave+8).
// B fetched once per k-step and reused by both M-tiles -> 4 WMMAs per
// (2 ds_load_b64 + 4 global_load_b32).
__device__ __forceinline__ void layer_gemm(const float* __restrict__ W,
                                           const float* __restrict__ bias,
                                           const float* ldsIn, float* ldsOut,
                                           bool do_relu, int lane, int wave)
{
    const int m16 = lane & 15;            // A row-in-tile / B,C column lane
    const int kg  = (lane >> 4) << 1;     // K group: 0 (lanes 0-15) or 2 (lanes 16-31)
    const int hi8 = (lane >> 4) << 3;     // C row offset: 0 or 8

    const int n0 = wave * 16 + m16;       // column for N-tile 0
    const int n1 = n0 + 128;              // column for N-tile 1 (nt = wave + 8)

    const float b0v = bias[n0];
    const float b1v = bias[n1];
    v8f c00 = {b0v, b0v, b0v, b0v, b0v, b0v, b0v, b0v};   // Mtile0 x Ntile0
    v8f c01 = {b1v, b1v, b1v, b1v, b1v, b1v, b1v, b1v};   // Mtile0 x Ntile1
    v8f c10 = c00;                                        // Mtile1 x Ntile0
    v8f c11 = c01;                                        // Mtile1 x Ntile1

    const float* __restrict__ Wp0 = W + (size_t)kg * 256 + n0;
    const float* __restrict__ Wp1 = Wp0 + 128;
    const float* Ap0 = ldsIn + m16 * LSTRIDE + kg;        // rows 0-15
    const float* Ap1 = Ap0 + 16 * LSTRIDE;                // rows 16-31

#pragma unroll 4
    for (int k0 = 0; k0 < 256; k0 += 4) {
        v2f a0 = *(const v2f*)(Ap0 + k0);
        v2f a1 = *(const v2f*)(Ap1 + k0);
        v2f b0, b1;
        b0.x = Wp0[(size_t)k0 * 256];
        b0.y = Wp0[(size_t)k0 * 256 + 256];
        b1.x = Wp1[(size_t)k0 * 256];
        b1.y = Wp1[(size_t)k0 * 256 + 256];
        c00 = __builtin_amdgcn_wmma_f32_16x16x4_f32(false, a0, false, b0,
                                                    (short)0, c00, false, false);
        c01 = __builtin_amdgcn_wmma_f32_16x16x4_f32(false, a0, false, b1,
                                                    (short)0, c01, false, false);
        c10 = __builtin_amdgcn_wmma_f32_16x16x4_f32(false, a1, false, b0,
                                                    (short)0, c10, false, false);
        c11 = __builtin_amdgcn_wmma_f32_16x16x4_f32(false, a1, false, b1,
                                                    (short)0, c11, false, false);
    }

#pragma unroll
    for (int r = 0; r < 8; ++r) {
        float v00 = c00[r], v01 = c01[r], v10 = c10[r], v11 = c11[r];
        if (do_relu) {
            v00 = fmaxf(v00, 0.f); v01 = fmaxf(v01, 0.f);
            v10 = fmaxf(v10, 0.f); v11 = fmaxf(v11, 0.f);
        }
        const int row0 = r + hi8;
        ldsOut[row0 * LSTRIDE + n0] = v00;
        ldsOut[row0 * LSTRIDE + n1] = v01;
        ldsOut[(row0 + 16) * LSTRIDE + n0] = v10;
        ldsOut[(row0 + 16) * LSTRIDE + n1] = v11;
    }
}

__global__ __launch_bounds__(256) void window_mlp_kernel(
    const float* __restrict__ x,
    const float* __restrict__ Wa, const float* __restrict__ ba,
    const float* __restrict__ We, const float* __restrict__ be,
    const float* __restrict__ Wr, const float* __restrict__ br,
    const float* __restrict__ Ws, const float* __restrict__ bs,
    float* __restrict__ out)
{
    __shared__ float buf0[MTILE * LSTRIDE];
    __shared__ float buf1[MTILE * LSTRIDE];
    __shared__ float attpart[MTILE * 17];
    __shared__ float attv[MTILE];

    const int t    = threadIdx.x;
    const int lane = t & 31;
    const int wave = t >> 5;

    // ---- stage 32 windows (32x256 fp32) into LDS; 2 rows per thread ----
#pragma unroll
    for (int rep = 0; rep < 2; ++rep) {
        const int w = (t >> 4) + rep * 16, r = t & 15;
        const int widx = blockIdx.x * MTILE + w;
        float4* dst = (float4*)&buf0[w * LSTRIDE + r * 16];
        if (widx < NWIN) {
            const int bb  = widx / (NH * NW);
            const int rem = widx % (NH * NW);
            const int wi = rem / NW, wj = rem % NW;
            const float* src = x + (size_t)bb * HW + (size_t)(wi * 8 + r) * 1024 + wj * 8;
            const float4* s4 = (const float4*)src;  // 32B-aligned (stride-8 windows)
            dst[0] = s4[0]; dst[1] = s4[1]; dst[2] = s4[2]; dst[3] = s4[3];
        } else {
            float4 z = make_float4(0.f, 0.f, 0.f, 0.f);
            dst[0] = z; dst[1] = z; dst[2] = z; dst[3] = z;
        }
    }
    __syncthreads();

    // ---- attention partials (overlapped with layer 1) ----
#pragma unroll
    for (int rep = 0; rep < 2; ++rep) {
        const int w = (t >> 4) + rep * 16, seg = t & 15;
        float p = 0.f;
#pragma unroll
        for (int i = 0; i < 16; ++i)
            p += buf0[w * LSTRIDE + seg * 16 + i] * Wa[seg * 16 + i];
        attpart[w * 17 + seg] = p;
    }

    // ---- layer 1: expanded = winf @ We + be ----
    layer_gemm(We, be, buf0, buf1, /*relu=*/false, lane, wave);
    __syncthreads();

    // ---- attention reduce: att[w] = relu(sum + ba) ----
    if (t < MTILE) {
        float s = ba[0];
#pragma unroll
        for (int i = 0; i < 16; ++i) s += attpart[t * 17 + i];
        attv[t] = fmaxf(s, 0.f);
    }

    // ---- layer 2: rec = relu(expanded @ Wr + br) ----
    layer_gemm(Wr, br, buf1, buf0, /*relu=*/true, lane, wave);
    __syncthreads();

    // ---- layer 3: rep = rec @ Ws + bs ----
    layer_gemm(Ws, bs, buf0, buf1, /*relu=*/false, lane, wave);
    __syncthreads();

    // ---- scatter-add: out += rep * att (overlapping windows -> atomics) ----
#pragma unroll
    for (int rep = 0; rep < 2; ++rep) {
        const int w = (t >> 4) + rep * 16, r = t & 15;
        const int widx = blockIdx.x * MTILE + w;
        if (widx < NWIN) {
            const int bb  = widx / (NH * NW);
            const int rem = widx % (NH * NW);
            const int wi = rem / NW, wj = rem % NW;
            const float a = attv[w];
            float* dst = out + (size_t)bb * HW + (size_t)(wi * 8 + r) * 1024 + wj * 8;
#pragma unroll
            for (int ci = 0; ci < 16; ++ci)
                atomicAdd(dst + ci, buf1[w * LSTRIDE + r * 16 + ci] * a);
        }
    }
}

__global__ __launch_bounds__(256) void copy_kernel(const float4* __restrict__ src,
                                                   float4* __restrict__ dst, int n4)
{
    int i = blockIdx.x * blockDim.x + threadIdx.x;
    if (i < n4) dst[i] = src[i];
}

extern "C" void kernel_launch(void* const* d_in, const int* in_sizes, int n_in,
                              void* d_out, int out_size, void* d_ws, size_t ws_size,
                              hipStream_t stream)
{
    (void)in_sizes; (void)n_in; (void)out_size; (void)d_ws; (void)ws_size;
    const float* x  = (const float*)d_in[0];
    const float* Wa = (const float*)d_in[1];
    const float* ba = (const float*)d_in[2];
    const float* We = (const float*)d_in[3];
    const float* be = (const float*)d_in[4];
    const float* Wr = (const float*)d_in[5];
    const float* br = (const float*)d_in[6];
    const float* Ws = (const float*)d_in[7];
    const float* bs = (const float*)d_in[8];
    float* out = (float*)d_out;

    // out = x  (base image; windows then atomically accumulate on top)
    const int n4 = NBATCH * HW / 4;  // 2,097,152 float4
    copy_kernel<<<(n4 + 255) / 256, 256, 0, stream>>>((const float4*)x, (float4*)out, n4);

    const int nblocks = (NWIN + MTILE - 1) / MTILE;  // 4033 M-tiles of 32 windows
    window_mlp_kernel<<<nblocks, 256, 0, stream>>>(x, Wa, ba, We, be, Wr, br, Ws, bs, out);
}